// DIST_loss_49331994362452
// MI455X (gfx1250) — compile-verified
//
#include <hip/hip_runtime.h>

typedef float v2f __attribute__((ext_vector_type(2)));
typedef float v4f __attribute__((ext_vector_type(4)));
typedef float v8f __attribute__((ext_vector_type(8)));

// Cross-lane wave32 sum on the matrix pipe.
// WMMA1: A[m][0]=acc(lane m), A[m][2]=acc(lane m+16), B=ones
//        -> D[m][n] = acc[m] + acc[m+16]
// Per-lane sum of the 8 D VGPRs collapses M (lanes 0-15 hold M=0..7,
// lanes 16-31 hold M=8..15), WMMA2 with ones-B fuses the two halves:
// every lane's d2[0] == full wave sum.
__device__ __forceinline__ float wave_sum_wmma(float acc) {
  v2f a;    a.x = acc;  a.y = 0.0f;
  v2f ones; ones.x = 1.0f; ones.y = 1.0f;
  v8f c = {};
  v8f d = __builtin_amdgcn_wmma_f32_16x16x4_f32(
      false, a, false, ones, (short)0, c, false, false);
  float h = ((d[0] + d[1]) + (d[2] + d[3])) + ((d[4] + d[5]) + (d[6] + d[7]));
  v2f a2; a2.x = h; a2.y = 0.0f;
  v8f d2 = __builtin_amdgcn_wmma_f32_16x16x4_f32(
      false, a2, false, ones, (short)0, c, false, false);
  return d2[0];
}

// Pass 1: stream pred/target with 128-bit nontemporal loads (2 rows / load),
// accumulate per-lane distances, WMMA wave reduce, LDS block reduce,
// one partial per block into d_ws.
__global__ void __launch_bounds__(256)
dist_partial_kernel(const float* __restrict__ pred,
                    const float* __restrict__ target,
                    float* __restrict__ partials,
                    long long n4, long long B) {
  const v4f* __restrict__ p4 = (const v4f*)pred;
  const v4f* __restrict__ t4 = (const v4f*)target;

  long long tid    = (long long)blockIdx.x * blockDim.x + threadIdx.x;
  long long stride = (long long)gridDim.x * blockDim.x;

  float acc = 0.0f;
  #pragma unroll 2
  for (long long i = tid; i < n4; i += stride) {
    if (i + stride < n4) {
      __builtin_prefetch(&p4[i + stride], 0, 0);
      __builtin_prefetch(&t4[i + stride], 0, 0);
    }
    v4f p = __builtin_nontemporal_load(&p4[i]);
    v4f t = __builtin_nontemporal_load(&t4[i]);
    float dx0 = p.x - t.x, dy0 = p.y - t.y;
    float dx1 = p.z - t.z, dy1 = p.w - t.w;
    acc += __builtin_amdgcn_sqrtf(dx0 * dx0 + dy0 * dy0);
    acc += __builtin_amdgcn_sqrtf(dx1 * dx1 + dy1 * dy1);
  }
  // odd-B tail (not hit for B = 8388608, kept for generality)
  if ((B & 1LL) && tid == 0) {
    float dx = pred[2 * (B - 1)]     - target[2 * (B - 1)];
    float dy = pred[2 * (B - 1) + 1] - target[2 * (B - 1) + 1];
    acc += __builtin_amdgcn_sqrtf(dx * dx + dy * dy);
  }

  // EXEC is all-ones here (full reconvergence after the loop) -> WMMA legal.
  float wave_total = wave_sum_wmma(acc);

  __shared__ float smem[8];  // 256 threads = 8 wave32s
  int lane = threadIdx.x & 31;
  int wave = threadIdx.x >> 5;
  if (lane == 0) smem[wave] = wave_total;
  __syncthreads();
  if (threadIdx.x == 0) {
    float s = 0.0f;
    #pragma unroll
    for (int w = 0; w < 8; ++w) s += smem[w];
    partials[blockIdx.x] = s;
  }
}

// Pass 2: deterministic fixed-order tree reduction of block partials,
// then divide by float(B+1) (2^23+1 is exact in f32, same as jnp f32 cast).
__global__ void __launch_bounds__(256)
dist_final_kernel(const float* __restrict__ partials, int n,
                  float* __restrict__ out, float denom) {
  __shared__ float smem[256];
  float s = 0.0f;
  for (int i = threadIdx.x; i < n; i += 256) s += partials[i];
  smem[threadIdx.x] = s;
  __syncthreads();
  #pragma unroll
  for (int off = 128; off > 0; off >>= 1) {
    if ((int)threadIdx.x < off) smem[threadIdx.x] += smem[threadIdx.x + off];
    __syncthreads();
  }
  if (threadIdx.x == 0) out[0] = smem[0] / denom;
}

extern "C" void kernel_launch(void* const* d_in, const int* in_sizes, int n_in,
                              void* d_out, int out_size, void* d_ws, size_t ws_size,
                              hipStream_t stream) {
  const float* pred   = (const float*)d_in[0];
  const float* target = (const float*)d_in[1];
  float* out          = (float*)d_out;
  float* partials     = (float*)d_ws;

  long long B  = (long long)in_sizes[0] / 2;  // rows
  long long n4 = B / 2;                       // float4 elements per array

  int block = 256;
  int grid  = 2048;  // 16K wave32s: saturates 23.3 TB/s HBM streaming
  long long maxPartials = (long long)(ws_size / sizeof(float));
  if (maxPartials > 0 && grid > maxPartials) grid = (int)maxPartials;
  if (grid < 1) grid = 1;

  dist_partial_kernel<<<grid, block, 0, stream>>>(pred, target, partials, n4, B);

  float denom = (float)(B + 1);  // f32-rounded, matching the reference
  dist_final_kernel<<<1, 256, 0, stream>>>(partials, grid, out, denom);
}